// NEXT_Simulator_33208687133353
// MI455X (gfx1250) — compile-verified
//
#include <hip/hip_runtime.h>
#include <math.h>

typedef __attribute__((ext_vector_type(2))) float v2f;
typedef __attribute__((ext_vector_type(8))) float v8f;

#define B_      2
#define NPART_  100
#define KDIFF_  100
#define E_      10000      // NPART_*KDIFF_
#define T_      550
#define TP_     560        // T padded to multiple of 16
#define S_      2209       // 47*47
#define NSIPM_  47
#define ST_TILES 139       // ceil(2209/16) -> 2224 padded rows
#define TGROUPS  5         // 5 waves * 7 tiles * 16 = 560 ticks
#define NT_PER   7
#define KC_      32        // K-chunk staged in LDS
#define GAUSS_NORM 3.9894228040143267f   // 1/(0.1*2.5066282746)

__device__ __forceinline__ float sigmoidf_(float x) {
    return 1.0f / (1.0f + expf(-x));
}

// ---------------------------------------------------------------------------
// Kernel 1: per-electron preprocessing: diffusion, lifetime weight, PMT MLP.
// Writes exy = x+y, ez, weight, and transposed padded pmtT[b][16][E].
// ---------------------------------------------------------------------------
__global__ void precompute_kernel(const float* __restrict__ pos,
                                  const float* __restrict__ disp,
                                  const float* __restrict__ randoms,
                                  const float* __restrict__ W1,
                                  const float* __restrict__ b1,
                                  const float* __restrict__ W2,
                                  const float* __restrict__ b2,
                                  const float* __restrict__ dscale,
                                  const float* __restrict__ pmt_scale,
                                  const float* __restrict__ lifetime_p,
                                  float* __restrict__ exy,
                                  float* __restrict__ ezv,
                                  float* __restrict__ wtv,
                                  float* __restrict__ pmtT)
{
    int i = blockIdx.x * blockDim.x + threadIdx.x;
    if (i >= B_ * NPART_ * KDIFF_) return;
    int b = i / (NPART_ * KDIFF_);
    int e = i % (NPART_ * KDIFF_);          // e = n*K + k (matches reshape)
    int n = e / KDIFF_;
    int k = e % KDIFF_;

    const float* p = pos + (size_t)(b * NPART_ + n) * 3;
    const float* d = disp + ((size_t)(b * NPART_ + n) * KDIFF_ + k) * 3;
    float sz = sqrtf(p[2]);
    float ex = dscale[0] * d[0] * sz + p[0];
    float ey = dscale[1] * d[1] * sz + p[1];
    float ez = dscale[2] * d[2] * sz + p[2];

    float r = randoms[(size_t)(b * NPART_ + n) * KDIFF_ + k];
    float w = sigmoidf_((expf(-ez / lifetime_p[0]) - r) * 1000.0f);

    // MLP: sigmoid([x,y]@W1 + b1) -> sigmoid(@W2 + b2) * pmt_scale
    float h1[28];
#pragma unroll
    for (int j = 0; j < 28; ++j)
        h1[j] = sigmoidf_(ex * W1[j] + ey * W1[28 + j] + b1[j]);
#pragma unroll
    for (int pch = 0; pch < 12; ++pch) {
        float acc = b2[pch];
#pragma unroll
        for (int j = 0; j < 28; ++j) acc += h1[j] * W2[j * 12 + pch];
        pmtT[((size_t)b * 16 + pch) * E_ + e] = sigmoidf_(acc) * pmt_scale[pch];
    }
#pragma unroll
    for (int pch = 12; pch < 16; ++pch)      // zero-pad rows 12..15
        pmtT[((size_t)b * 16 + pch) * E_ + e] = 0.0f;

    int be = b * E_ + e;
    exy[be] = ex + ey;
    ezv[be] = ez;
    wtv[be] = w;
}

// ---------------------------------------------------------------------------
// Kernel 2: Gaussian tick response, stored K-PAIR-INTERLEAVED:
//   expvP[((b*(E/2)+e/2)*TP + t)*2 + (e&1)]
// so a WMMA B-fragment (two adjacent K rows, same tick) is one b64 load.
// Thread mapping (parity fastest) keeps the stores perfectly coalesced.
// ---------------------------------------------------------------------------
__global__ void expv_kernel(const float* __restrict__ ezv,
                            const float* __restrict__ wtv,
                            float* __restrict__ expvP)
{
    long i = (long)blockIdx.x * blockDim.x + threadIdx.x;
    if (i >= (long)B_ * E_ * TP_) return;
    int  r  = (int)(i & 1);
    long j  = i >> 1;
    int  t  = (int)(j % TP_);
    long bp = j / TP_;                       // b*(E/2) + pair
    int  b  = (int)(bp / (E_ / 2));
    int  pr = (int)(bp % (E_ / 2));
    long be = (long)b * E_ + 2 * pr + r;

    float val = 0.0f;
    if (t < T_) {
        float dt = ((float)t + 0.5f) - ezv[be];
        val = wtv[be] * GAUSS_NORM * expf(-dt * dt * 10.0f);
    }
    expvP[i] = val;
}

// ---------------------------------------------------------------------------
// Kernel 3: PMT GEMM  C(16 x 560) = pmtT(16 x E) * expv(E x 560), one wave
// per (batch, 16-tick tile), full-K loop with V_WMMA_F32_16X16X4_F32.
// ---------------------------------------------------------------------------
__global__ void pmt_wmma_kernel(const float* __restrict__ pmtT,
                                const float* __restrict__ expvP,
                                float* __restrict__ out_pmt)
{
    int blk = blockIdx.x;                 // b*35 + nt
    int b   = blk / 35;
    int n0  = (blk % 35) * 16;
    int lane = threadIdx.x & 31;
    int m  = lane & 15;
    int hk = lane >> 4;

    const float* A  = pmtT  + (size_t)b * 16 * E_;
    const float* Bp = expvP + (size_t)b * (E_ / 2) * TP_ * 2;

    v8f c = {};
    for (int k0 = 0; k0 < E_; k0 += 4) {
        int kk = k0 + 2 * hk;             // even
        v2f a, bb;
        a.x = A[(size_t)m * E_ + kk];     // contiguous pair -> b64
        a.y = A[(size_t)m * E_ + kk + 1];
        bb  = *(const v2f*)&Bp[((size_t)(kk >> 1) * TP_ + n0 + m) * 2];
        c = __builtin_amdgcn_wmma_f32_16x16x4_f32(false, a, false, bb,
                                                  (short)0, c, false, false);
    }
#pragma unroll
    for (int rrow = 0; rrow < 8; ++rrow) {
        int M = rrow + 8 * hk;
        int Ncol = n0 + m;
        if (M < 12 && Ncol < T_)
            out_pmt[((size_t)b * 12 + M) * T_ + Ncol] = c[rrow];
    }
}

// ---------------------------------------------------------------------------
// Kernel 4: SiPM GEMM  C(2209 x 550) = sipm_resp^T(2209 x E) * expv(E x 560).
// Block = 5 waves sharing one 16-row SiPM strip; wave w owns tick group
// [w*112, w*112+112). The A tile (KC x 16 exps) is computed ONCE per K-chunk
// into LDS by all 160 threads (5x fewer transcendentals), fragments read via
// ds_load_b64; B fragments are single b64 loads from pair-interleaved expv.
// ---------------------------------------------------------------------------
__global__ void __launch_bounds__(32 * TGROUPS)
sipm_wmma_kernel(const float* __restrict__ exy,
                 const float* __restrict__ expvP,
                 const float* __restrict__ si_scale,
                 const float* __restrict__ sigma_p,
                 float* __restrict__ out_sipm)
{
    __shared__ float a_lds[KC_ / 2][16][2];   // [kpair][m][parity]
    __shared__ float ssum_s[16];
    __shared__ float scale_s[16];

    int blk = blockIdx.x;
    int b   = blk / ST_TILES;
    int s0  = (blk % ST_TILES) * 16;
    int tid  = threadIdx.x;                   // 0..159
    int wave = tid >> 5;                      // T-group
    int lane = tid & 31;
    int m  = lane & 15;
    int hk = lane >> 4;
    int n0 = wave * (NT_PER * 16);

    if (tid < 16) {                           // per-row constants
        int s = s0 + tid;
        float ss = 0.0f, sc = 0.0f;
        if (s < S_) {
            int si = s / NSIPM_, sj = s % NSIPM_;
            ss = 10.0f * (float)(si + sj) - 470.0f;  // (-235+10i)+(-235+10j)
            sc = si_scale[s];
        }
        ssum_s[tid]  = ss;
        scale_s[tid] = sc;
    }
    float sg = sigma_p[0];
    float inv_s2 = 1.0f / (sg * sg);
    const float* Ex = exy   + b * E_;
    const float* Bp = expvP + (size_t)b * (E_ / 2) * TP_ * 2;

    v8f c[NT_PER] = {};
    __syncthreads();

    for (int k0 = 0; k0 < E_; k0 += KC_) {
        // ---- cooperative A tile: KC_ x 16 responses into LDS ----
        for (int idx = tid; idx < KC_ * 16; idx += 32 * TGROUPS) {
            int kl = idx >> 4;                // 0..KC_-1
            int mm = idx & 15;
            float d = Ex[k0 + kl] - ssum_s[mm];
            a_lds[kl >> 1][mm][kl & 1] = __expf(-d * d * inv_s2) * scale_s[mm];
        }
        __syncthreads();

        if (k0 + KC_ < E_)                    // warm next chunk's B rows
            __builtin_prefetch(Bp + ((size_t)((k0 + KC_) >> 1) * TP_ + n0 + m) * 2, 0, 1);

        // ---- WMMA phase: 8 K-steps x 7 N-tiles ----
#pragma unroll
        for (int ks = 0; ks < KC_; ks += 4) {
            int kk = ks + 2 * hk;             // even
            v2f a = *(const v2f*)&a_lds[kk >> 1][m][0];           // ds_load_b64
            const float* Brow = Bp + (size_t)((k0 + kk) >> 1) * TP_ * 2;
#pragma unroll
            for (int nt = 0; nt < NT_PER; ++nt) {
                int n = n0 + nt * 16 + m;
                v2f bb = *(const v2f*)&Brow[n * 2];               // b64 load
                c[nt] = __builtin_amdgcn_wmma_f32_16x16x4_f32(false, a, false, bb,
                                                              (short)0, c[nt],
                                                              false, false);
            }
        }
        __syncthreads();
    }

#pragma unroll
    for (int nt = 0; nt < NT_PER; ++nt) {
        int Ncol = n0 + nt * 16 + m;
#pragma unroll
        for (int rrow = 0; rrow < 8; ++rrow) {
            int srow = s0 + rrow + 8 * hk;
            if (srow < S_ && Ncol < T_)
                out_sipm[((size_t)b * S_ + srow) * T_ + Ncol] = c[nt][rrow];
        }
    }
}

// ---------------------------------------------------------------------------
extern "C" void kernel_launch(void* const* d_in, const int* in_sizes, int n_in,
                              void* d_out, int out_size, void* d_ws, size_t ws_size,
                              hipStream_t stream)
{
    const float* positions     = (const float*)d_in[0];
    const float* displacements = (const float*)d_in[1];
    const float* randoms       = (const float*)d_in[2];
    const float* W1            = (const float*)d_in[3];
    const float* b1            = (const float*)d_in[4];
    const float* W2            = (const float*)d_in[5];
    const float* b2            = (const float*)d_in[6];
    const float* dscale        = (const float*)d_in[7];
    const float* sigma         = (const float*)d_in[8];
    const float* pmt_scale     = (const float*)d_in[9];
    const float* si_scale      = (const float*)d_in[10];
    const float* lifetime      = (const float*)d_in[11];

    // Workspace layout (floats): exy | ez | w | pmtT(2*16*E) | expvP(2*E*560)
    float* ws    = (float*)d_ws;
    float* exy   = ws;
    float* ezv   = ws + 20000;
    float* wtv   = ws + 40000;
    float* pmtT  = ws + 60000;        // 320000 floats
    float* expvP = ws + 380000;       // 11,200,000 floats (~44.8 MB, L2-resident)

    float* out_pmt  = (float*)d_out;               // (2,12,550)
    float* out_sipm = (float*)d_out + 2 * 12 * T_; // (2,2209,550)

    precompute_kernel<<<(B_ * NPART_ * KDIFF_ + 255) / 256, 256, 0, stream>>>(
        positions, displacements, randoms, W1, b1, W2, b2,
        dscale, pmt_scale, lifetime, exy, ezv, wtv, pmtT);

    long tot = (long)B_ * E_ * TP_;
    expv_kernel<<<(int)((tot + 255) / 256), 256, 0, stream>>>(ezv, wtv, expvP);

    pmt_wmma_kernel<<<B_ * 35, 32, 0, stream>>>(pmtT, expvP, out_pmt);

    sipm_wmma_kernel<<<B_ * ST_TILES, 32 * TGROUPS, 0, stream>>>(
        exy, expvP, si_scale, sigma, out_sipm);
}